// HybridDecomposition_47734266527858
// MI455X (gfx1250) — compile-verified
//
#include <hip/hip_runtime.h>
#include <stdint.h>

#ifndef __has_builtin
#define __has_builtin(x) 0
#endif

#if __has_builtin(__builtin_amdgcn_wmma_f32_16x16x4_f32)
#define USE_WMMA 1
#else
#define USE_WMMA 0
#endif

#if __has_builtin(__builtin_amdgcn_tensor_load_to_lds) && __has_builtin(__builtin_amdgcn_s_wait_tensorcnt)
#define USE_TDM 1
#else
#define USE_TDM 0
#endif

#define S_LEN 4096
#define C_CH  128
#define B_N   32
#define K_WIN 25
#define PADW  12
#define L_CHUNK 64
#define ROWS (L_CHUNK + 2*PADW)      /* 88 rows staged per chunk */
#define CHUNKS (S_LEN / L_CHUNK)     /* 64 */
#define CPB 4                        /* chunks per block */
#define BLOCK_CHUNKS (CHUNKS / CPB)  /* 16 */
#define NSTAT 7
#define EPS_F 1e-8f

typedef __attribute__((ext_vector_type(2))) float        v2f;
typedef __attribute__((ext_vector_type(8))) float        v8f;
typedef __attribute__((ext_vector_type(4))) unsigned int v4u;
typedef __attribute__((ext_vector_type(8))) int          v8i;
typedef __attribute__((ext_vector_type(4))) int          v4i;

// ---------------------------------------------------------------------------
// Kernel 1: per-(b, 4-chunk) block. TDM-stage x rows (with halo, zero-padded)
// into LDS, compute trend via WMMA f32 16x16x4 band-matrix chain, store trend,
// accumulate 7 per-(b,c) moment sums into deterministic per-block partials.
// ---------------------------------------------------------------------------
__global__ __launch_bounds__(128) void k1_trend_moments(
    const float* __restrict__ x, float* __restrict__ trend_out,
    float* __restrict__ partials) {
  __shared__ float smem[ROWS * C_CH];                 // 45056 B

  const int b     = blockIdx.x >> 4;                  // / BLOCK_CHUNKS
  const int cb    = blockIdx.x & (BLOCK_CHUNKS - 1);
  const int tid   = threadIdx.x;
  const int w     = tid >> 5;                         // wave id 0..3
  const int lane  = tid & 31;
  const int lhalf = lane >> 4;                        // 0 / 1
  const int lm    = lane & 15;

#if USE_WMMA
  // A_j: constant 16x4 slices of banded 16x40 matrix W (W[m][k]=1/25 iff m<=k<=m+24)
  // A layout (f32 16x4): lanes 0-15 M=lane K={4j,4j+1}; lanes 16-31 M=lane-16 K={4j+2,4j+3}
  v2f A[10];
#pragma unroll
  for (int j = 0; j < 10; ++j) {
    const int k0 = 4 * j + 2 * lhalf;
    const float a0 = (k0     >= lm && k0     <= lm + 24) ? 0.04f : 0.0f;
    const float a1 = (k0 + 1 >= lm && k0 + 1 <= lm + 24) ? 0.04f : 0.0f;
    A[j].x = a0; A[j].y = a1;
  }
#endif

  float acc[2][NSTAT];
#pragma unroll
  for (int g = 0; g < 2; ++g)
#pragma unroll
    for (int s = 0; s < NSTAT; ++s) acc[g][s] = 0.0f;

  for (int sc = 0; sc < CPB; ++sc) {
    const int chunk = cb * CPB + sc;
    const int n0 = chunk * L_CHUNK;

    // head zero-padding (only chunk 0 has rows before t=0)
    if (chunk == 0) {
#pragma unroll
      for (int r = 0; r < PADW; ++r) smem[r * C_CH + tid] = 0.0f;
    }

    int rowStart  = n0 - PADW;
    int ldsRowOff = 0;
    if (rowStart < 0) { ldsRowOff = -rowStart; rowStart = 0; }
    const int loadRows = ROWS - ldsRowOff;
    const unsigned remRows = (unsigned)(S_LEN - rowStart);  // tail OOB rows -> zero
    const unsigned long long gaddr =
        (unsigned long long)(uintptr_t)(x + (long long)(b * S_LEN + rowStart) * C_CH);

#if USE_TDM
    if (w == 0) {
      const unsigned ldsAddr =
          (unsigned)(uintptr_t)&smem[0] + (unsigned)(ldsRowOff * C_CH * 4);
      v4u g0;
      g0.x = 1u;                                           // count=1, load descriptor
      g0.y = ldsAddr;                                      // lds_addr [63:32]
      g0.z = (unsigned)(gaddr & 0xFFFFFFFFull);            // global_addr lo
      g0.w = (unsigned)((gaddr >> 32) & 0x01FFFFFFull) | 0x80000000u; // addr hi | type=2
      v8i g1;
      g1[0] = (int)(2u << 16);                             // data_size = 4 B
      g1[1] = (int)(((unsigned)C_CH) << 16);               // tensor_dim0 = 128 (bits 63:48)
      g1[2] = (int)((remRows & 0xFFFFu) << 16);            // tensor_dim1 lo16 (bits 95:80)
      g1[3] = (int)((remRows >> 16) | (((unsigned)C_CH) << 16)); // dim1 hi | tile_dim0=128
      g1[4] = loadRows;                                    // tile_dim1
      g1[5] = C_CH;                                        // tensor_dim0_stride = 128
      g1[6] = 0;
      g1[7] = 0;
      v4i g2 = {0, 0, 0, 0};
      v4i g3 = {0, 0, 0, 0};
      v8i g4 = {0, 0, 0, 0, 0, 0, 0, 0};
      __builtin_amdgcn_tensor_load_to_lds(g0, g1, g2, g3, g4, 0);
      __builtin_amdgcn_s_wait_tensorcnt(0);
    }
#else
    for (int r = 0; r < loadRows; ++r) {
      float v = (r < (int)remRows)
                    ? x[(long long)(b * S_LEN + rowStart + r) * C_CH + tid]
                    : 0.0f;
      smem[(ldsRowOff + r) * C_CH + tid] = v;
    }
#endif
    __syncthreads();

    // wave w owns channel groups {2w, 2w+1}; tiles are 16 time x 16 chan
#pragma unroll
    for (int cgi = 0; cgi < 2; ++cgi) {
      const int cg  = 2 * w + cgi;
      const int col = cg * 16 + lm;
      for (int tt = 0; tt < L_CHUNK / 16; ++tt) {
        v8f d = {0.f, 0.f, 0.f, 0.f, 0.f, 0.f, 0.f, 0.f};
#if USE_WMMA
#pragma unroll
        for (int j = 0; j < 10; ++j) {
          // B layout (f32 4x16): lanes 0-15 hold K={4j,4j+1}, lanes 16-31 K={4j+2,4j+3}; N=lane&15
          const int r0 = tt * 16 + 4 * j + 2 * lhalf;
          v2f bv;
          bv.x = smem[r0 * C_CH + col];
          bv.y = smem[(r0 + 1) * C_CH + col];
          d = __builtin_amdgcn_wmma_f32_16x16x4_f32(false, A[j], false, bv,
                                                    (short)0, d, false, false);
        }
#else
#pragma unroll
        for (int r = 0; r < 8; ++r) {
          const int m = r + 8 * lhalf;
          float s = 0.0f;
          for (int k = 0; k < K_WIN; ++k) s += smem[(tt * 16 + m + k) * C_CH + col];
          d[r] = s * 0.04f;
        }
#endif
        // D layout: VGPR r -> time m = r + 8*lhalf, channel = col. Store + moments.
        const int tbase = n0 + tt * 16;
#pragma unroll
        for (int r = 0; r < 8; ++r) {
          const int m   = r + 8 * lhalf;
          const float t = d[r];
          const float xv = smem[(tt * 16 + m + PADW) * C_CH + col];
          const float sg = (r & 1) ? -1.0f : 1.0f;   // (-1)^n, n parity == r parity
          acc[cgi][0] += xv;
          acc[cgi][1] += xv * xv;
          acc[cgi][2] += sg * xv;
          acc[cgi][3] += t;
          acc[cgi][4] += t * t;
          acc[cgi][5] += sg * t;
          acc[cgi][6] += xv * t;
          trend_out[(b * S_LEN + tbase + m) * C_CH + col] = t;
        }
      }
    }
    __syncthreads();   // before next chunk's stage overwrites LDS
  }

  // combine lane pairs (lane L holds times 0..7 slice, L+16 holds 8..15 slice of same channel)
#pragma unroll
  for (int cgi = 0; cgi < 2; ++cgi) {
    const int cg = 2 * w + cgi;
#pragma unroll
    for (int st = 0; st < NSTAT; ++st) {
      float v = acc[cgi][st];
      v += __shfl_down(v, 16, 32);
      if (lane < 16) {
        partials[((b * BLOCK_CHUNKS + cb) * 8 + st) * C_CH + cg * 16 + lane] = v;
      }
    }
  }
}

// ---------------------------------------------------------------------------
// Kernel 2: deterministic combine of partials; closed-form variances -> cond.
// ---------------------------------------------------------------------------
__global__ __launch_bounds__(128) void k2_cond(const float* __restrict__ partials,
                                               float* __restrict__ condf) {
  const int b = blockIdx.x;
  const int c = threadIdx.x;
  float s[NSTAT];
#pragma unroll
  for (int st = 0; st < NSTAT; ++st) s[st] = 0.0f;
  for (int g = 0; g < BLOCK_CHUNKS; ++g)
#pragma unroll
    for (int st = 0; st < NSTAT; ++st)
      s[st] += partials[((b * BLOCK_CHUNKS + g) * 8 + st) * C_CH + c];

  const float Sx = s[0], Sxx = s[1], Bt = s[2], St = s[3],
              Stt = s[4], Sat = s[5], Sxt = s[6];
  const float N = 4096.0f, Nm1 = 4095.0f;

  // var(seasonal0 + resid) = var(x - t)
  const float sum1 = Sx - St;
  const float sq1  = Sxx - 2.0f * Sxt + Stt;
  const float var_xt = (sq1 - sum1 * sum1 / N) / Nm1;
  // var(trend + resid) = var(0.5x - u), u[n] = (-1)^n * Bt/(2N)
  const float sum2 = 0.5f * Sx;
  const float sq2  = 0.25f * Sxx - Bt * Bt / (4.0f * N);
  const float var_tr = (sq2 - sum2 * sum2 / N) / Nm1;
  // var(resid) = var(0.5x - t - u)
  const float sum3 = 0.5f * Sx - St;
  const float sq3  = 0.25f * Sxx - Sxt + Stt
                   - (Bt / N) * (0.5f * Bt - Sat) + Bt * Bt / (4.0f * N);
  const float var_res = (sq3 - sum3 * sum3 / N) / Nm1;

  const float d_t = 1.0f - var_res / (var_tr + EPS_F);
  const float d_s = 1.0f - var_res / (var_xt + EPS_F);
  condf[b * C_CH + c] = (d_t > 0.5f || d_s > 0.5f) ? 1.0f : 0.0f;
}

// ---------------------------------------------------------------------------
// Kernel 3: seasonal = cond ? x/(trend+eps) : x - trend   (float4 streaming)
// ---------------------------------------------------------------------------
__global__ __launch_bounds__(256) void k3_seasonal(
    const float* __restrict__ x, const float* __restrict__ trend,
    const float* __restrict__ condf, float* __restrict__ seas) {
  const long long i = ((long long)blockIdx.x * 256 + threadIdx.x) * 4;
  const int c = (int)(i & 127);
  const int b = (int)(i >> 19);     // i / (4096*128)
  const float4 xv = *(const float4*)(x + i);
  const float4 tv = *(const float4*)(trend + i);
  const float4 cf = *(const float4*)(condf + b * C_CH + c);
  float4 o;
  o.x = (cf.x != 0.0f) ? xv.x / (tv.x + EPS_F) : xv.x - tv.x;
  o.y = (cf.y != 0.0f) ? xv.y / (tv.y + EPS_F) : xv.y - tv.y;
  o.z = (cf.z != 0.0f) ? xv.z / (tv.z + EPS_F) : xv.z - tv.z;
  o.w = (cf.w != 0.0f) ? xv.w / (tv.w + EPS_F) : xv.w - tv.w;
  *(float4*)(seas + i) = o;
}

extern "C" void kernel_launch(void* const* d_in, const int* in_sizes, int n_in,
                              void* d_out, int out_size, void* d_ws, size_t ws_size,
                              hipStream_t stream) {
  (void)in_sizes; (void)n_in; (void)out_size; (void)ws_size;
  const float* x = (const float*)d_in[0];
  const long long TOT = (long long)B_N * S_LEN * C_CH;   // 16,777,216
  float* trend = (float*)d_out;
  float* seas  = (float*)d_out + TOT;
  float* partials = (float*)d_ws;                        // 32*16*8*128 = 2 MB
  float* condf = partials + (long long)B_N * BLOCK_CHUNKS * 8 * C_CH;

  k1_trend_moments<<<B_N * BLOCK_CHUNKS, 128, 0, stream>>>(x, trend, partials);
  k2_cond<<<B_N, 128, 0, stream>>>(partials, condf);
  k3_seasonal<<<(int)(TOT / 4 / 256), 256, 0, stream>>>(x, trend, condf, seas);
}